// SPNNeuronBig_8478265442321
// MI455X (gfx1250) — compile-verified
//
#include <hip/hip_runtime.h>
#include <cstdint>

#define F_IN 10000
#define NROWS 4096
#define SPLITS 10
#define COLS_PER_SPLIT 1000
#define GROUPS_PER_SPLIT 250   // float4 column groups per split
#define ROWTILE 16
#define PSTRIDE 12             // partials per row, padded from 10 -> 12 for WMMA K

typedef __attribute__((ext_vector_type(2))) float v2f;
typedef __attribute__((ext_vector_type(8))) float v8f;

// log2-domain 3-component Gaussian-mixture log-likelihood.
// P layout (per column): [0..2]=mu, [4..6]=inv_sigma*sqrt(0.5/ln2), [8..10]=log2-consts
__device__ __forceinline__ float lse2_3(float x, const float* P) {
  float z0 = (x - P[0]) * P[4];
  float z1 = (x - P[1]) * P[5];
  float z2 = (x - P[2]) * P[6];
  float b0 = __builtin_fmaf(-z0, z0, P[8]);
  float b1 = __builtin_fmaf(-z1, z1, P[9]);
  float b2 = __builtin_fmaf(-z2, z2, P[10]);
  float m  = fmaxf(fmaxf(b0, b1), b2);
  float s  = __builtin_amdgcn_exp2f(b0 - m)
           + __builtin_amdgcn_exp2f(b1 - m)
           + __builtin_amdgcn_exp2f(b2 - m);
  return m + __builtin_amdgcn_logf(s);   // v_log_f32 == log2
}

// One-time per-launch: fold log_softmax(w1), ln(sigma), 0.5*ln(2pi) and the
// 1/ln2 rescaling into per-feature constants, scattered through rand_idxs so
// the main kernel reads x in natural (coalesced) column order.
__global__ __launch_bounds__(256)
void spn_setup(const float* __restrict__ means, const float* __restrict__ stds,
               const float* __restrict__ w1, const int* __restrict__ ridx,
               float* __restrict__ pc, float* __restrict__ partial) {
  int i = blockIdx.x * 256 + threadIdx.x;
  if (i < F_IN) {
    float a0 = w1[i * 3 + 0], a1 = w1[i * 3 + 1], a2 = w1[i * 3 + 2];
    float mx  = fmaxf(fmaxf(a0, a1), a2);
    float lse = mx + logf(expf(a0 - mx) + expf(a1 - mx) + expf(a2 - mx));
    const float INVLN2  = 1.4426950408889634f;
    const float HLOG2PI = 0.9189385332046727f;          // 0.5*ln(2*pi)
    const float KS      = 0.8493218002880191f;          // sqrt(0.5/ln2)
    int d = ridx[i];
    float* o = pc + (size_t)d * 12;
    #pragma unroll
    for (int m = 0; m < 3; ++m) {
      float mu = means[i * 3 + m];
      float sd = stds[i * 3 + m];
      o[m]     = mu;
      o[4 + m] = KS / sd;
      o[8 + m] = ((w1[i * 3 + m] - lse) - logf(sd) - HLOG2PI) * INVLN2;
    }
    o[3] = 0.f; o[7] = 0.f; o[11] = 0.f;
  }
  if (i < NROWS) {  // zero the K-padding of the partials array each launch
    partial[(size_t)i * PSTRIDE + 10] = 0.f;
    partial[(size_t)i * PSTRIDE + 11] = 0.f;
  }
}

// Main streaming kernel: block = 16 rows x 1000 columns. Params for the
// column split are async-copied into LDS; x streams coalesced as float4.
__global__ __launch_bounds__(256)
void spn_main(const float* __restrict__ x, const float* __restrict__ pc,
              float* __restrict__ partial) {
  __shared__ float spc[COLS_PER_SPLIT * 12];   // 48000 B of params
  __shared__ float sred[ROWTILE * 256];        // 16384 B reduction scratch

  const int t     = threadIdx.x;
  const int row0  = blockIdx.x * ROWTILE;
  const int split = blockIdx.y;

  // ---- CDNA5 async global->LDS staging of this split's parameters ----
  const float4* pg = reinterpret_cast<const float4*>(pc)
                   + (size_t)split * (COLS_PER_SPLIT * 3);
  unsigned lds0 = (unsigned)(unsigned long long)(&spc[0]);
  for (int i = t; i < COLS_PER_SPLIT * 3; i += 256) {
    const float4* src = pg + i;
    unsigned dst = lds0 + 16u * (unsigned)i;
    asm volatile("global_load_async_to_lds_b128 %0, %1, off"
                 :: "v"(dst), "v"(src) : "memory");
  }
  asm volatile("s_wait_asynccnt 0x0" ::: "memory");
  __syncthreads();

  float acc[ROWTILE];
  #pragma unroll
  for (int r = 0; r < ROWTILE; ++r) acc[r] = 0.f;

  if (t < GROUPS_PER_SPLIT) {
    float P[48];                      // params for this thread's 4 columns
    #pragma unroll
    for (int i = 0; i < 48; ++i) P[i] = spc[48 * t + i];

    const float4* x4 = reinterpret_cast<const float4*>(x);
    size_t base = (size_t)row0 * (F_IN / 4)
                + (size_t)split * GROUPS_PER_SPLIT + t;
    #pragma unroll
    for (int r = 0; r < ROWTILE; ++r) {
      float4 xv = x4[base + (size_t)r * (F_IN / 4)];
      acc[r] = lse2_3(xv.x, &P[0])  + lse2_3(xv.y, &P[12])
             + lse2_3(xv.z, &P[24]) + lse2_3(xv.w, &P[36]);
    }
  }

  #pragma unroll
  for (int r = 0; r < ROWTILE; ++r) sred[r * 256 + t] = acc[r];
  __syncthreads();

  // Each of the 8 waves reduces 2 rows: 8 strided LDS reads + shfl_xor tree.
  const int w = t >> 5, l = t & 31;
  #pragma unroll
  for (int rr = 0; rr < 2; ++rr) {
    int r = w * 2 + rr;
    float s = 0.f;
    #pragma unroll
    for (int k = 0; k < 8; ++k) s += sred[r * 256 + l + k * 32];
    #pragma unroll
    for (int m = 16; m >= 1; m >>= 1) s += __shfl_xor(s, m, 32);
    if (l == 0) partial[(size_t)(row0 + r) * PSTRIDE + split] = s;
  }
}

// Final reduction of the 10 (padded to 12) per-row split partials via
// v_wmma_f32_16x16x4_f32 with a ones-B matrix: D = A x 1 + C sums K=4 per op,
// chained 3x through the accumulator. One wave handles 16 rows.
__global__ __launch_bounds__(32)
void spn_reduce(const float* __restrict__ partial, float* __restrict__ out) {
  const int l    = threadIdx.x;
  const int row0 = blockIdx.x * 16;
  const int m    = l & 15;      // A-matrix row
  const int kh   = l >> 4;      // lanes 16-31 hold K=2,3
  v8f c = {};
  v2f b; b[0] = 1.f; b[1] = 1.f;
  #pragma unroll
  for (int ck = 0; ck < 3; ++ck) {
    const float* p = partial + (size_t)(row0 + m) * PSTRIDE + ck * 4 + kh * 2;
    v2f a; a[0] = p[0]; a[1] = p[1];
    c = __builtin_amdgcn_wmma_f32_16x16x4_f32(false, a, false, b,
                                              (short)0, c, false, false);
  }
  // C/D layout: lane n in [0,16) holds rows 0..7 in vgprs 0..7 (col n);
  // lane n in [16,32) holds rows 8..15. Convert log2 -> ln at the end.
  const float LN2 = 0.6931471805599453f;
  if (l == 0) {
    #pragma unroll
    for (int r = 0; r < 8; ++r) out[row0 + r] = c[r] * LN2;
  } else if (l == 16) {
    #pragma unroll
    for (int r = 0; r < 8; ++r) out[row0 + 8 + r] = c[r] * LN2;
  }
}

extern "C" void kernel_launch(void* const* d_in, const int* in_sizes, int n_in,
                              void* d_out, int out_size, void* d_ws, size_t ws_size,
                              hipStream_t stream) {
  const float* x     = (const float*)d_in[0];
  const float* means = (const float*)d_in[1];
  const float* stds  = (const float*)d_in[2];
  const float* w1    = (const float*)d_in[3];
  const int*   ridx  = (const int*)d_in[8];
  float* out = (float*)d_out;

  float* pc      = (float*)d_ws;                        // 10000*12*4 = 480000 B
  float* partial = (float*)((char*)d_ws + 480000);      // 4096*12*4  = 196608 B

  spn_setup<<<40, 256, 0, stream>>>(means, stds, w1, ridx, pc, partial);
  spn_main<<<dim3(NROWS / ROWTILE, SPLITS), 256, 0, stream>>>(x, pc, partial);
  spn_reduce<<<NROWS / 16, 32, 0, stream>>>(partial, out);
}